// GCN_29867202576799
// MI455X (gfx1250) — compile-verified
//
#include <hip/hip_runtime.h>

typedef __attribute__((ext_vector_type(2))) float v2f;
typedef __attribute__((ext_vector_type(8))) float v8f;

#define NNODES 100000
#define NEDGES 3200000
#define FIN 128
#define FHID 16
#define FOUT 2

static inline int cdiv(long long a, long long b) { return (int)((a + b - 1) / b); }

// ---- init: deg = 1 (self-loop), accumulators = 0 (ws is poisoned by harness) ----
__global__ void k_init(float* __restrict__ deg, float* __restrict__ acc1,
                       float* __restrict__ acc2, int n) {
  int t = blockIdx.x * blockDim.x + threadIdx.x;
  if (t < n * FHID) acc1[t] = 0.0f;
  if (t < n * FOUT) acc2[t] = 0.0f;
  if (t < n)        deg[t]  = 1.0f;
}

// ---- in-degree (incl. self-loop via init) ----
__global__ void k_degree(const long long* __restrict__ dst, float* __restrict__ deg, int ne) {
  int e = blockIdx.x * blockDim.x + threadIdx.x;
  if (e < ne) atomicAdd(&deg[(int)dst[e]], 1.0f);
}

// ---- deg -> deg^{-1/2} in place (deg >= 1 always) ----
__global__ void k_rsqrt(float* __restrict__ deg, int n) {
  int i = blockIdx.x * blockDim.x + threadIdx.x;
  if (i < n) deg[i] = rsqrtf(deg[i]);
}

// ---- h_scaled[i,:] = dis[i] * (x[i,:] @ W1), full-precision fp32 WMMA ----
// One wave per 16-row tile; K loop = 32 x V_WMMA_F32_16X16X4_F32.
__global__ void __launch_bounds__(256)
k_gemm1_wmma(const float* __restrict__ x, const float* __restrict__ W1,
             const float* __restrict__ dis, float* __restrict__ hs, int n) {
  __shared__ float sW[FIN * FHID];  // 8 KB of 320 KB LDS
  for (int i = threadIdx.x; i < FIN * FHID; i += blockDim.x) sW[i] = W1[i];
  __syncthreads();

  const int lane = threadIdx.x & 31;
  const int wave = blockIdx.x * (blockDim.x >> 5) + (threadIdx.x >> 5);
  const int row0 = wave * 16;
  if (row0 >= n) return;                 // wave-uniform: EXEC stays all-1s for WMMA

  const int m  = lane & 15;              // tile row (A) / column (B,C)
  const int kh = (lane >> 4) << 1;       // lanes 16-31 carry K+2,K+3

  const float* xr = x + (size_t)(row0 + m) * FIN + kh;
  v8f c = {};
  #pragma unroll
  for (int k0 = 0; k0 < FIN; k0 += 4) {
    v2f a = *(const v2f*)(xr + k0);      // A 16x4 f32: 2 dwords/lane
    v2f b;                                // B 4x16 f32 from LDS
    b.x = sW[(k0 + kh + 0) * FHID + m];
    b.y = sW[(k0 + kh + 1) * FHID + m];
    c = __builtin_amdgcn_wmma_f32_16x16x4_f32(false, a, false, b, (short)0, c,
                                              false, false);
  }
  // C/D layout: vgpr r -> M = r (lanes 0-15) / r+8 (lanes 16-31), N = m.
  #pragma unroll
  for (int r = 0; r < 8; ++r) {
    int row = row0 + ((lane < 16) ? r : r + 8);
    hs[(size_t)row * FHID + m] = c[r] * dis[row];
  }
}

// ---- layer-1 scatter: acc1[dst, f] += h_scaled[src, f]; f in low 4 bits ->
//      16 gathers = one coalesced row, 16 atomics = one 64B L2 line ----
__global__ void k_scatter16(const long long* __restrict__ src, const long long* __restrict__ dst,
                            const float* __restrict__ hs, float* __restrict__ acc,
                            long long total) {
  long long t = (long long)blockIdx.x * blockDim.x + threadIdx.x;
  if (t >= total) return;
  int e = (int)(t >> 4), f = (int)(t & 15);
  int s = (int)src[e], d = (int)dst[e];
  atomicAdd(&acc[(size_t)d * FHID + f], hs[(size_t)s * FHID + f]);
}

// ---- z = relu(dis[dst]*(edge_sum + self_loop) + b1), in place over acc1 ----
__global__ void k_post1(const float* __restrict__ dis, const float* __restrict__ hs,
                        float* __restrict__ acc1, const float* __restrict__ b1, int n) {
  int t = blockIdx.x * blockDim.x + threadIdx.x;
  if (t >= n * FHID) return;
  int i = t >> 4, f = t & 15;
  float v = dis[i] * (acc1[t] + hs[t]) + b1[f];
  acc1[t] = fmaxf(v, 0.0f);
}

// ---- g_scaled = dis[i] * (z[i,:] @ W2)   (16 -> 2, trivial FLOPs) ----
__global__ void k_gemm2(const float* __restrict__ z, const float* __restrict__ W2,
                        const float* __restrict__ dis, float* __restrict__ gs, int n) {
  int i = blockIdx.x * blockDim.x + threadIdx.x;
  if (i >= n) return;
  float g0 = 0.0f, g1 = 0.0f;
  #pragma unroll
  for (int k = 0; k < FHID; ++k) {
    float zk = z[(size_t)i * FHID + k];
    g0 = fmaf(zk, W2[k * FOUT + 0], g0);
    g1 = fmaf(zk, W2[k * FOUT + 1], g1);
  }
  float di = dis[i];
  gs[(size_t)i * FOUT + 0] = g0 * di;
  gs[(size_t)i * FOUT + 1] = g1 * di;
}

// ---- layer-2 scatter (2 floats/edge) ----
__global__ void k_scatter2(const long long* __restrict__ src, const long long* __restrict__ dst,
                           const float* __restrict__ gs, float* __restrict__ acc,
                           long long total) {
  long long t = (long long)blockIdx.x * blockDim.x + threadIdx.x;
  if (t >= total) return;
  int e = (int)(t >> 1), f = (int)(t & 1);
  int s = (int)src[e], d = (int)dst[e];
  atomicAdd(&acc[(size_t)d * FOUT + f], gs[(size_t)s * FOUT + f]);
}

// ---- out = dis[dst]*(edge_sum + self_loop) + b2 ----
__global__ void k_final(const float* __restrict__ dis, const float* __restrict__ gs,
                        const float* __restrict__ acc2, const float* __restrict__ b2,
                        float* __restrict__ out, int n) {
  int t = blockIdx.x * blockDim.x + threadIdx.x;
  if (t >= n * FOUT) return;
  int i = t >> 1, f = t & 1;
  out[t] = dis[i] * (acc2[t] + gs[t]) + b2[f];
}

extern "C" void kernel_launch(void* const* d_in, const int* in_sizes, int n_in,
                              void* d_out, int out_size, void* d_ws, size_t ws_size,
                              hipStream_t stream) {
  const float*     x  = (const float*)d_in[0];
  const long long* ei = (const long long*)d_in[1];  // int64 [2, NEDGES]
  const float*     W1 = (const float*)d_in[2];      // [128,16]
  const float*     b1 = (const float*)d_in[3];      // [16]
  const float*     W2 = (const float*)d_in[4];      // [16,2]
  const float*     b2 = (const float*)d_in[5];      // [2]
  float* out = (float*)d_out;                       // [N,2]

  const long long* src = ei;
  const long long* dst = ei + NEDGES;

  // workspace layout: 37*N floats (~14.8 MB)
  float* ws   = (float*)d_ws;
  float* deg  = ws;                               // N   (becomes dis in place)
  float* hs   = deg  + NNODES;                    // N*16 (dis[src]-scaled h)
  float* acc1 = hs   + (size_t)NNODES * FHID;     // N*16 (becomes relu output)
  float* gs   = acc1 + (size_t)NNODES * FHID;     // N*2
  float* acc2 = gs   + (size_t)NNODES * FOUT;     // N*2

  const int B = 256;  // 8 waves/block on wave32
  k_init    <<<cdiv((long long)NNODES * FHID, B), B, 0, stream>>>(deg, acc1, acc2, NNODES);
  k_degree  <<<cdiv(NEDGES, B), B, 0, stream>>>(dst, deg, NEDGES);
  k_rsqrt   <<<cdiv(NNODES, B), B, 0, stream>>>(deg, NNODES);
  k_gemm1_wmma<<<cdiv(cdiv(NNODES, 16), B / 32), B, 0, stream>>>(x, W1, deg, hs, NNODES);
  k_scatter16<<<cdiv((long long)NEDGES * FHID, B), B, 0, stream>>>(src, dst, hs, acc1,
                                                                   (long long)NEDGES * FHID);
  k_post1   <<<cdiv((long long)NNODES * FHID, B), B, 0, stream>>>(deg, hs, acc1, b1, NNODES);
  k_gemm2   <<<cdiv(NNODES, B), B, 0, stream>>>(acc1, W2, deg, gs, NNODES);
  k_scatter2<<<cdiv((long long)NEDGES * FOUT, B), B, 0, stream>>>(src, dst, gs, acc2,
                                                                  (long long)NEDGES * FOUT);
  k_final   <<<cdiv((long long)NNODES * FOUT, B), B, 0, stream>>>(deg, gs, acc2, b2, out, NNODES);
}